// IERF_71975061946417
// MI455X (gfx1250) — compile-verified
//
#include <hip/hip_runtime.h>

// ierf(z) = 0.5 * (z*erf(z) + (exp(-z^2) + z*sqrt(pi) - 1)/sqrt(pi))
// with A&S 7.1.26 erf approximation. Elementwise over 2^27 f32 -> HBM-bound.
// Strategy: b128 non-temporal global loads/stores, v_exp_f32/v_rcp_f32
// transcendentals co-executing with FMA chain, 2x float4 per thread for ILP.

typedef float v4f __attribute__((ext_vector_type(4)));

__device__ __forceinline__ float ierf1(float z) {
    const float A1 =  0.254829592f;
    const float A2 = -0.284496736f;
    const float A3 =  1.421413741f;
    const float A4 = -1.453152027f;
    const float A5 =  1.061405429f;
    const float T_C = 0.3275911f;
    const float INV_RAD_PI = 0.5641895835477563f;   // 1/sqrt(pi)
    const float NEG_LOG2E  = -1.4426950408889634f;  // -log2(e)

    float az   = __builtin_fabsf(z);
    // exp(-z*z) == exp2(-z*z*log2(e))  -> v_exp_f32 (TRANS32)
    float fexp = __builtin_amdgcn_exp2f(z * z * NEG_LOG2E);
    // t = 1/(1 + T_C*|z|) -> v_rcp_f32 (TRANS32)
    float t    = __builtin_amdgcn_rcpf(__builtin_fmaf(T_C, az, 1.0f));
    float poly = __builtin_fmaf(t, A5, A4);
    poly       = __builtin_fmaf(t, poly, A3);
    poly       = __builtin_fmaf(t, poly, A2);
    poly       = __builtin_fmaf(t, poly, A1);
    // ferf = sign(z) * (1 - t*poly*fexp)
    float ferf = __builtin_copysignf(__builtin_fmaf(-(t * poly), fexp, 1.0f), z);
    // 0.5*(z*ferf + (fexp-1)/sqrt(pi) + z)
    float r = __builtin_fmaf(z, ferf,
              __builtin_fmaf(fexp - 1.0f, INV_RAD_PI, z));
    return 0.5f * r;
}

__device__ __forceinline__ v4f ierf4(v4f v) {
    v4f r;
    r.x = ierf1(v.x);
    r.y = ierf1(v.y);
    r.z = ierf1(v.z);
    r.w = ierf1(v.w);
    return r;
}

// Each thread processes 2 float4s (8 elements), spaced blockDim apart so each
// wave32 issues 512B-contiguous b128 transactions. Non-temporal hints keep the
// 1GB read-once/write-once stream from thrashing the 192MB L2.
__global__ __launch_bounds__(256) void ierf_kernel(const float* __restrict__ x,
                                                   float* __restrict__ y,
                                                   int n4, int ntail) {
    const v4f* __restrict__ xv = (const v4f*)x;
    v4f* __restrict__ yv = (v4f*)y;

    int tid  = threadIdx.x;
    int base = blockIdx.x * 512 + tid;  // in float4 units; two tiles of 256

    int i0 = base;
    int i1 = base + 256;

    // Two independent load->compute->store chains for ILP; compiler will
    // front-load both global_load_b128 th:NT before the first s_wait_loadcnt.
    if (i0 < n4) {
        v4f a = __builtin_nontemporal_load(xv + i0);
        if (i1 < n4) {
            v4f b = __builtin_nontemporal_load(xv + i1);
            __builtin_nontemporal_store(ierf4(a), yv + i0);
            __builtin_nontemporal_store(ierf4(b), yv + i1);
        } else {
            __builtin_nontemporal_store(ierf4(a), yv + i0);
        }
    }

    // Scalar tail (n % 4 elements) handled by block 0.
    if (blockIdx.x == 0 && tid < ntail) {
        int i = n4 * 4 + tid;
        y[i] = ierf1(x[i]);
    }
}

extern "C" void kernel_launch(void* const* d_in, const int* in_sizes, int n_in,
                              void* d_out, int out_size, void* d_ws, size_t ws_size,
                              hipStream_t stream) {
    const float* x = (const float*)d_in[0];
    float* y = (float*)d_out;
    int n = in_sizes[0];          // 32*4096*1024 = 134,217,728
    int n4 = n >> 2;              // float4 count
    int ntail = n & 3;

    int blocks = (n4 + 511) / 512;  // 512 float4s per block (256 thr x 2)
    if (blocks < 1) blocks = 1;     // still run tail if n < 4

    ierf_kernel<<<blocks, 256, 0, stream>>>(x, y, n4, ntail);
}